// DyHeadBlock_9156870275200
// MI455X (gfx1250) — compile-verified
//
#include <hip/hip_runtime.h>

typedef float v2f __attribute__((ext_vector_type(2)));
typedef float v8f __attribute__((ext_vector_type(8)));

#define C_CH 256
#define NB 2

// ---------------------------------------------------------------------------
// Offset conv: 3x3, 256 -> 27.  Channels 0..17 -> offset (raw), 18..26 ->
// sigmoid -> mask.  One block per (b, pixel); patch staged in LDS.
// ---------------------------------------------------------------------------
__global__ void offconv_kernel(const float* __restrict__ x,
                               const float* __restrict__ ww,   // 27 x 256 x 3 x 3
                               const float* __restrict__ wb,   // 27
                               float* __restrict__ offmap,     // [B][18][H][W]
                               float* __restrict__ maskmap,    // [B][9][H][W]
                               int H, int W) {
    __shared__ float patch[2304];
    int HW = H * W;
    int b  = blockIdx.x / HW;
    int p  = blockIdx.x % HW;
    int oy = p / W, ox = p % W;

    for (int i = threadIdx.x; i < 2304; i += blockDim.x) {
        int c = i / 9, j = i % 9;
        int y = oy + j / 3 - 1, xx = ox + j % 3 - 1;
        float v = 0.f;
        if (y >= 0 && y < H && xx >= 0 && xx < W)
            v = x[((size_t)(b * C_CH + c) * H + y) * W + xx];
        patch[i] = v;
    }
    __syncthreads();

    int o = threadIdx.x;
    if (o < 27) {
        const float* wr = ww + (size_t)o * 2304;
        float acc = wb[o];
        for (int i = 0; i < 2304; ++i) acc += patch[i] * wr[i];
        if (o < 18)
            offmap[((size_t)(b * 18 + o) * H + oy) * W + ox] = acc;
        else
            maskmap[((size_t)(b * 9 + (o - 18)) * H + oy) * W + ox] =
                1.f / (1.f + expf(-acc));
    }
}

// ---------------------------------------------------------------------------
// Fused modulated-deformable-conv GEMM with f32 WMMA.
// Block = 512 threads (16 waves). Each block: 16 output pixels of one (b).
// Wave w computes output channels [16w, 16w+16) via V_WMMA_F32_16X16X4_F32.
// A = weight tile (O x K), B = sampled tile (K x 16px) staged in LDS.
// K ordering = c*9 + tap, matching weight.reshape(O, C, 9).
// ---------------------------------------------------------------------------
__global__ __launch_bounds__(512)
void dcn_wmma_kernel(const float* __restrict__ x, int Hin, int Win,
                     const float* __restrict__ offmap,
                     const float* __restrict__ maskmap,
                     int Hoff, int Woff, int offstep,
                     int stride, int Ho, int Wo,
                     const float* __restrict__ wk,   // 256 x 2304 (O major)
                     float* __restrict__ out) {      // [B][256][Ho*Wo]
    __shared__ int   s_idx[9 * 16 * 4];
    __shared__ float s_w  [9 * 16 * 4];
    __shared__ float s_samp[288 * 16];   // 32ch * 9tap  x  16px

    int HWo    = Ho * Wo;
    int ntiles = HWo >> 4;
    int b    = blockIdx.x / ntiles;
    int tile = blockIdx.x % ntiles;
    int t    = threadIdx.x;

    // ---- per-(pixel,tap) bilinear metadata ----
    if (t < 144) {
        int k = t / 16, n = t % 16;
        int p  = tile * 16 + n;
        int oy = p / Wo, ox = p % Wo;
        int oyo = oy * offstep, oxo = ox * offstep;
        float offy = offmap[((size_t)(b * 18 + 2 * k)     * Hoff + oyo) * Woff + oxo];
        float offx = offmap[((size_t)(b * 18 + 2 * k + 1) * Hoff + oyo) * Woff + oxo];
        float m    = maskmap[((size_t)(b * 9 + k)         * Hoff + oyo) * Woff + oxo];
        float py  = (float)(oy * stride - 1 + k / 3) + offy;
        float pxf = (float)(ox * stride - 1 + k % 3) + offx;
        float y0f = floorf(py), x0f = floorf(pxf);
        float ly = py - y0f, lx = pxf - x0f;
        int y0 = (int)y0f, x0 = (int)x0f;
        for (int c4 = 0; c4 < 4; ++c4) {
            int yi = y0 + (c4 >> 1), xi = x0 + (c4 & 1);
            float w = ((c4 >> 1) ? ly : 1.f - ly) * ((c4 & 1) ? lx : 1.f - lx) * m;
            if (!(yi >= 0 && yi < Hin && xi >= 0 && xi < Win)) w = 0.f;
            int yc = min(max(yi, 0), Hin - 1), xc = min(max(xi, 0), Win - 1);
            s_idx[(k * 16 + n) * 4 + c4] = yc * Win + xc;
            s_w  [(k * 16 + n) * 4 + c4] = w;
        }
    }
    __syncthreads();

    int wave = t >> 5;
    int lane = t & 31;
    int half = lane >> 4;     // K sub-block select (ISA 32-bit A/B layout)
    int ln   = lane & 15;     // M for A, N for B
    int obase = wave * 16;

    v8f acc = {0.f, 0.f, 0.f, 0.f, 0.f, 0.f, 0.f, 0.f};

    for (int cc = 0; cc < C_CH; cc += 32) {
        __syncthreads();
        // fill sampled K-chunk: rows = c_local*9 + tap, cols = 16 pixels
        for (int i = t; i < 32 * 144; i += 512) {
            int cl  = i / 144;
            int rem = i % 144;                   // tap*16 + px
            const float* xp = x + (size_t)(b * C_CH + cc + cl) * Hin * Win;
            const int*   ip = &s_idx[rem * 4];
            const float* wp = &s_w[rem * 4];
            float v = wp[0] * xp[ip[0]] + wp[1] * xp[ip[1]] +
                      wp[2] * xp[ip[2]] + wp[3] * xp[ip[3]];
            int k = rem >> 4, n = rem & 15;
            s_samp[(cl * 9 + k) * 16 + n] = v;
        }
        __syncthreads();

        const float* wbase = wk + (size_t)(obase + ln) * 2304 + cc * 9;
        for (int kk = 0; kk < 288; kk += 4) {
            int ka = kk + (half ? 2 : 0);
            v2f a, bm;
            a.x  = wbase[ka];
            a.y  = wbase[ka + 1];
            bm.x = s_samp[ka * 16 + ln];
            bm.y = s_samp[(ka + 1) * 16 + ln];
            acc = __builtin_amdgcn_wmma_f32_16x16x4_f32(
                false, a, false, bm, (short)0, acc, false, false);
        }
    }

    // D layout: VGPR j -> M = j + 8*half, N = ln
    int p = tile * 16 + ln;
    for (int j = 0; j < 8; ++j) {
        int o = obase + j + half * 8;
        out[(size_t)(b * C_CH + o) * HWo + p] = acc[j];
    }
}

// ---------------------------------------------------------------------------
// GroupNorm, in-place.  grid = (16 groups, B), block = 256.
// ---------------------------------------------------------------------------
__global__ void gn_kernel(float* __restrict__ xb,
                          const float* __restrict__ gamma,
                          const float* __restrict__ beta, int HW) {
    const int CG = C_CH / 16;                 // 16 channels per group
    int g = blockIdx.x, b = blockIdx.y;
    size_t base = ((size_t)b * C_CH + g * CG) * HW;
    int n = CG * HW;
    float s = 0.f, ss = 0.f;
    for (int i = threadIdx.x; i < n; i += blockDim.x) {
        float v = xb[base + i];
        s += v; ss += v * v;
    }
    __shared__ float rs[256], rss[256];
    rs[threadIdx.x] = s; rss[threadIdx.x] = ss;
    __syncthreads();
    for (int st = 128; st > 0; st >>= 1) {
        if (threadIdx.x < st) {
            rs[threadIdx.x]  += rs[threadIdx.x + st];
            rss[threadIdx.x] += rss[threadIdx.x + st];
        }
        __syncthreads();
    }
    float mu  = rs[0] / n;
    float var = rss[0] / n - mu * mu;
    float inv = rsqrtf(var + 1e-5f);
    for (int i = threadIdx.x; i < n; i += blockDim.x) {
        int c = g * CG + i / HW;
        xb[base + i] = (xb[base + i] - mu) * inv * gamma[c] + beta[c];
    }
}

// ---------------------------------------------------------------------------
// Align-corners bilinear resize (B*C = 512 planes).
// ---------------------------------------------------------------------------
__global__ void resize_kernel(const float* __restrict__ in, float* __restrict__ o,
                              int Hi, int Wi, int Ho, int Wo) {
    int total = NB * C_CH * Ho * Wo;
    float sy = (float)(Hi - 1) / (float)(Ho - 1);
    float sx = (float)(Wi - 1) / (float)(Wo - 1);
    for (int i = blockIdx.x * blockDim.x + threadIdx.x; i < total;
         i += gridDim.x * blockDim.x) {
        int p  = i % (Ho * Wo);
        int bc = i / (Ho * Wo);
        int oy = p / Wo, ox = p % Wo;
        float ys = oy * sy, xs = ox * sx;
        int y0 = (int)floorf(ys); int y1 = min(y0 + 1, Hi - 1);
        int x0 = (int)floorf(xs); int x1 = min(x0 + 1, Wi - 1);
        float wy = ys - y0, wx = xs - x0;
        const float* ib = in + (size_t)bc * Hi * Wi;
        float a  = ib[y0 * Wi + x0] * (1.f - wy) + ib[y1 * Wi + x0] * wy;
        float bb = ib[y0 * Wi + x1] * (1.f - wy) + ib[y1 * Wi + x1] * wy;
        o[i] = a * (1.f - wx) + bb * wx;
    }
}

// ---------------------------------------------------------------------------
// Global average pool: grid = B*C planes, block = 256.
// ---------------------------------------------------------------------------
__global__ void gap_kernel(const float* __restrict__ in, float* __restrict__ gap,
                           int HW) {
    const float* p = in + (size_t)blockIdx.x * HW;
    float s = 0.f;
    for (int i = threadIdx.x; i < HW; i += blockDim.x) s += p[i];
    __shared__ float r[256];
    r[threadIdx.x] = s;
    __syncthreads();
    for (int st = 128; st > 0; st >>= 1) {
        if (threadIdx.x < st) r[threadIdx.x] += r[threadIdx.x + st];
        __syncthreads();
    }
    if (threadIdx.x == 0) gap[blockIdx.x] = r[0] / (float)HW;
}

// scale attention scalar per (feat, batch): hsigmoid(relu(gap . sw + sb))
__global__ void attn_kernel(const float* __restrict__ gap,  // [f][B][256]
                            const float* __restrict__ sw,
                            const float* __restrict__ sb,
                            float* __restrict__ attn) {      // [f][B]
    int f = blockIdx.x, b = blockIdx.y;
    float s = gap[(size_t)(f * NB + b) * C_CH + threadIdx.x] * sw[threadIdx.x];
    __shared__ float r[256];
    r[threadIdx.x] = s;
    __syncthreads();
    for (int st = 128; st > 0; st >>= 1) {
        if (threadIdx.x < st) r[threadIdx.x] += r[threadIdx.x + st];
        __syncthreads();
    }
    if (threadIdx.x == 0) {
        float a = fmaxf(r[0] + sb[0], 0.f);
        attn[f * NB + b] = fminf(fmaxf((a + 3.f) / 6.f, 0.f), 1.f);
    }
}

__global__ void combine_kernel(const float* __restrict__ f0,
                               const float* __restrict__ f1,
                               const float* __restrict__ f2,
                               const float* __restrict__ attn, int nf,
                               float* __restrict__ mf, int HW) {
    int total = NB * C_CH * HW;
    float invn = 1.f / (float)nf;
    for (int i = blockIdx.x * blockDim.x + threadIdx.x; i < total;
         i += gridDim.x * blockDim.x) {
        int b = i / (C_CH * HW);
        float s = f0[i] * attn[0 * NB + b] + f1[i] * attn[1 * NB + b];
        if (nf == 3) s += f2[i] * attn[2 * NB + b];
        mf[i] = s * invn;
    }
}

// DyReLU MLP: 256 -> 64 (relu) -> 1024 (hsigmoid).  grid = B.
__global__ void dyrelu_mlp_kernel(const float* __restrict__ gap,
                                  const float* __restrict__ w1,
                                  const float* __restrict__ b1,
                                  const float* __restrict__ w2,
                                  const float* __restrict__ b2,
                                  float* __restrict__ coef) {
    int b = blockIdx.x, t = threadIdx.x;
    __shared__ float g[256];
    __shared__ float h[64];
    g[t] = gap[b * C_CH + t];
    __syncthreads();
    if (t < 64) {
        float s = b1[t];
        const float* wr = w1 + (size_t)t * 256;
        for (int c = 0; c < 256; ++c) s += g[c] * wr[c];
        h[t] = fmaxf(s, 0.f);
    }
    __syncthreads();
    for (int o = t; o < 1024; o += 256) {
        float s = b2[o];
        const float* wr = w2 + (size_t)o * 64;
        for (int c = 0; c < 64; ++c) s += h[c] * wr[c];
        coef[b * 1024 + o] = fminf(fmaxf((s + 3.f) / 6.f, 0.f), 1.f);
    }
}

__global__ void dyrelu_apply_kernel(const float* __restrict__ mf,
                                    const float* __restrict__ coef,
                                    float* __restrict__ o, int HW) {
    int total = NB * C_CH * HW;
    for (int i = blockIdx.x * blockDim.x + threadIdx.x; i < total;
         i += gridDim.x * blockDim.x) {
        int b = i / (C_CH * HW);
        int c = (i / HW) % C_CH;
        const float* cf = coef + b * 1024;
        float a1  = (cf[c]       - 0.5f) * 2.f + 1.f;
        float b1_ =  cf[256 + c] - 0.5f;
        float a2  = (cf[512 + c] - 0.5f) * 2.f;
        float b2_ =  cf[768 + c] - 0.5f;
        float v = mf[i];
        o[i] = fmaxf(v * a1 + b1_, v * a2 + b2_);
    }
}

// ---------------------------------------------------------------------------
extern "C" void kernel_launch(void* const* d_in, const int* in_sizes, int n_in,
                              void* d_out, int out_size, void* d_ws, size_t ws_size,
                              hipStream_t stream) {
    const float* x0     = (const float*)d_in[0];
    const float* x1     = (const float*)d_in[1];
    const float* x2     = (const float*)d_in[2];
    const float* off_w  = (const float*)d_in[3];
    const float* off_b  = (const float*)d_in[4];
    const float* w_mid  = (const float*)d_in[5];
    const float* g_mid  = (const float*)d_in[6];
    const float* be_mid = (const float*)d_in[7];
    const float* w_low  = (const float*)d_in[8];
    const float* g_low  = (const float*)d_in[9];
    const float* be_low = (const float*)d_in[10];
    const float* w_high = (const float*)d_in[11];
    const float* g_high = (const float*)d_in[12];
    const float* be_high= (const float*)d_in[13];
    const float* sw     = (const float*)d_in[14];
    const float* sb     = (const float*)d_in[15];
    const float* dy1_w  = (const float*)d_in[16];
    const float* dy1_b  = (const float*)d_in[17];
    const float* dy2_w  = (const float*)d_in[18];
    const float* dy2_b  = (const float*)d_in[19];
    float* out = (float*)d_out;

    const int HW0 = 6400, HW1 = 1600, HW2 = 400;

    float* W = (float*)d_ws;
    size_t cur = 0;
    auto alloc = [&](size_t n) { float* p = W + cur; cur += n; return p; };
    float* off0 = alloc((size_t)NB * 18 * HW0);
    float* off1 = alloc((size_t)NB * 18 * HW1);
    float* off2 = alloc((size_t)NB * 18 * HW2);
    float* msk0 = alloc((size_t)NB * 9 * HW0);
    float* msk1 = alloc((size_t)NB * 9 * HW1);
    float* msk2 = alloc((size_t)NB * 9 * HW2);
    float* mid0 = alloc((size_t)NB * C_CH * HW0);
    float* hi0r = alloc((size_t)NB * C_CH * HW1);
    float* hi0  = alloc((size_t)NB * C_CH * HW0);
    float* mid1 = alloc((size_t)NB * C_CH * HW1);
    float* low1 = alloc((size_t)NB * C_CH * HW1);
    float* hi1r = alloc((size_t)NB * C_CH * HW2);
    float* hi1  = alloc((size_t)NB * C_CH * HW1);
    float* mid2 = alloc((size_t)NB * C_CH * HW2);
    float* low2 = alloc((size_t)NB * C_CH * HW2);
    float* mf0  = alloc((size_t)NB * C_CH * HW0);
    float* mf1  = alloc((size_t)NB * C_CH * HW1);
    float* mf2  = alloc((size_t)NB * C_CH * HW2);
    float* gapb = alloc(3 * NB * C_CH);
    float* gap2 = alloc(NB * C_CH);
    float* attn = alloc(8);
    float* coef = alloc(NB * 1024);

    // 1) offset conv per level
    offconv_kernel<<<NB * HW0, 128, 0, stream>>>(x0, off_w, off_b, off0, msk0, 80, 80);
    offconv_kernel<<<NB * HW1, 128, 0, stream>>>(x1, off_w, off_b, off1, msk1, 40, 40);
    offconv_kernel<<<NB * HW2, 128, 0, stream>>>(x2, off_w, off_b, off2, msk2, 20, 20);

    // 2) the seven DCN GEMMs (WMMA)
    dcn_wmma_kernel<<<NB * (HW0 / 16), 512, 0, stream>>>(x0, 80, 80, off0, msk0, 80, 80, 1, 1, 80, 80, w_mid,  mid0);
    dcn_wmma_kernel<<<NB * (HW1 / 16), 512, 0, stream>>>(x1, 40, 40, off0, msk0, 80, 80, 2, 1, 40, 40, w_high, hi0r);
    dcn_wmma_kernel<<<NB * (HW1 / 16), 512, 0, stream>>>(x1, 40, 40, off1, msk1, 40, 40, 1, 1, 40, 40, w_mid,  mid1);
    dcn_wmma_kernel<<<NB * (HW1 / 16), 512, 0, stream>>>(x0, 80, 80, off1, msk1, 40, 40, 1, 2, 40, 40, w_low,  low1);
    dcn_wmma_kernel<<<NB * (HW2 / 16), 512, 0, stream>>>(x2, 20, 20, off1, msk1, 40, 40, 2, 1, 20, 20, w_high, hi1r);
    dcn_wmma_kernel<<<NB * (HW2 / 16), 512, 0, stream>>>(x2, 20, 20, off2, msk2, 20, 20, 1, 1, 20, 20, w_mid,  mid2);
    dcn_wmma_kernel<<<NB * (HW2 / 16), 512, 0, stream>>>(x1, 40, 40, off2, msk2, 20, 20, 1, 2, 20, 20, w_low,  low2);

    // 3) GroupNorm (in-place, before any resize)
    dim3 gng(16, NB);
    gn_kernel<<<gng, 256, 0, stream>>>(mid0, g_mid,  be_mid,  HW0);
    gn_kernel<<<gng, 256, 0, stream>>>(hi0r, g_high, be_high, HW1);
    gn_kernel<<<gng, 256, 0, stream>>>(mid1, g_mid,  be_mid,  HW1);
    gn_kernel<<<gng, 256, 0, stream>>>(low1, g_low,  be_low,  HW1);
    gn_kernel<<<gng, 256, 0, stream>>>(hi1r, g_high, be_high, HW2);
    gn_kernel<<<gng, 256, 0, stream>>>(mid2, g_mid,  be_mid,  HW2);
    gn_kernel<<<gng, 256, 0, stream>>>(low2, g_low,  be_low,  HW2);

    // 4) resize high features up one level
    resize_kernel<<<(NB * C_CH * HW0 + 255) / 256, 256, 0, stream>>>(hi0r, hi0, 40, 40, 80, 80);
    resize_kernel<<<(NB * C_CH * HW1 + 255) / 256, 256, 0, stream>>>(hi1r, hi1, 20, 20, 40, 40);

    // 5) per-level attention combine + DyReLU
    // Level 0: [mid0, hi0]
    gap_kernel<<<NB * C_CH, 256, 0, stream>>>(mid0, gapb + 0 * NB * C_CH, HW0);
    gap_kernel<<<NB * C_CH, 256, 0, stream>>>(hi0,  gapb + 1 * NB * C_CH, HW0);
    attn_kernel<<<dim3(2, NB), 256, 0, stream>>>(gapb, sw, sb, attn);
    combine_kernel<<<(NB * C_CH * HW0 + 255) / 256, 256, 0, stream>>>(mid0, hi0, nullptr, attn, 2, mf0, HW0);
    gap_kernel<<<NB * C_CH, 256, 0, stream>>>(mf0, gap2, HW0);
    dyrelu_mlp_kernel<<<NB, 256, 0, stream>>>(gap2, dy1_w, dy1_b, dy2_w, dy2_b, coef);
    dyrelu_apply_kernel<<<(NB * C_CH * HW0 + 255) / 256, 256, 0, stream>>>(mf0, coef, out, HW0);

    // Level 1: [mid1, low1, hi1]
    gap_kernel<<<NB * C_CH, 256, 0, stream>>>(mid1, gapb + 0 * NB * C_CH, HW1);
    gap_kernel<<<NB * C_CH, 256, 0, stream>>>(low1, gapb + 1 * NB * C_CH, HW1);
    gap_kernel<<<NB * C_CH, 256, 0, stream>>>(hi1,  gapb + 2 * NB * C_CH, HW1);
    attn_kernel<<<dim3(3, NB), 256, 0, stream>>>(gapb, sw, sb, attn);
    combine_kernel<<<(NB * C_CH * HW1 + 255) / 256, 256, 0, stream>>>(mid1, low1, hi1, attn, 3, mf1, HW1);
    gap_kernel<<<NB * C_CH, 256, 0, stream>>>(mf1, gap2, HW1);
    dyrelu_mlp_kernel<<<NB, 256, 0, stream>>>(gap2, dy1_w, dy1_b, dy2_w, dy2_b, coef);
    dyrelu_apply_kernel<<<(NB * C_CH * HW1 + 255) / 256, 256, 0, stream>>>(
        mf1, coef, out + (size_t)NB * C_CH * HW0, HW1);

    // Level 2: [mid2, low2]
    gap_kernel<<<NB * C_CH, 256, 0, stream>>>(mid2, gapb + 0 * NB * C_CH, HW2);
    gap_kernel<<<NB * C_CH, 256, 0, stream>>>(low2, gapb + 1 * NB * C_CH, HW2);
    attn_kernel<<<dim3(2, NB), 256, 0, stream>>>(gapb, sw, sb, attn);
    combine_kernel<<<(NB * C_CH * HW2 + 255) / 256, 256, 0, stream>>>(mid2, low2, nullptr, attn, 2, mf2, HW2);
    gap_kernel<<<NB * C_CH, 256, 0, stream>>>(mf2, gap2, HW2);
    dyrelu_mlp_kernel<<<NB, 256, 0, stream>>>(gap2, dy1_w, dy1_b, dy2_w, dy2_b, coef);
    dyrelu_apply_kernel<<<(NB * C_CH * HW2 + 255) / 256, 256, 0, stream>>>(
        mf2, coef, out + (size_t)NB * C_CH * (HW0 + HW1), HW2);
}